// LocLoss_65635690217943
// MI455X (gfx1250) — compile-verified
//
#include <hip/hip_runtime.h>
#include <math.h>

// LocLoss: per-sample argmax over the (192x192) cls map -> gather loc at the
// argmax -> smooth-L1 vs closed-form bias -> mean over 2*b elements.
// Bandwidth-bound: ~36 MB of cls streaming (~1.6 us at 23.3 TB/s).  No matrix
// structure exists in this computation, so no WMMA by design; CDNA5-specific
// paths: wave32 shuffle reductions, clustered global_load_b128 streaming.

#define LOC_H 192
#define LOC_W 192
#define BLOCK_THREADS 1024            // 32 wave32s -> one well-fed WGP
#define STEP (BLOCK_THREADS * 4)      // elements covered per unrolled iter

// Cross-thread argmax combine: max value, min index on ties (matches
// jnp.argmax first-occurrence semantics).  Lexicographic max on (v, -i) is an
// associative+commutative semilattice, so reduction order cannot change it.
__device__ __forceinline__ void amax_combine(float& v, int& i, float ov, int oi) {
    if (ov > v || (ov == v && oi < i)) { v = ov; i = oi; }
}

// Shared tail: wave32 xor-tree -> LDS -> wave0 xor-tree -> lane0 computes the
// per-sample smooth-L1 pair and writes it to the workspace.
__device__ __forceinline__ void block_finish(float best, int bidx, int b, int hw,
                                             const float* __restrict__ loc,
                                             const float* __restrict__ center_rate,
                                             float* __restrict__ per_sample) {
    const int tid  = threadIdx.x;
    const int lane = tid & 31;
    const int wave = tid >> 5;

    #pragma unroll
    for (int off = 16; off > 0; off >>= 1) {
        const float ov = __shfl_xor(best, off, 32);
        const int   oi = __shfl_xor(bidx, off, 32);
        amax_combine(best, bidx, ov, oi);
    }

    __shared__ float s_v[BLOCK_THREADS / 32];
    __shared__ int   s_i[BLOCK_THREADS / 32];
    if (lane == 0) { s_v[wave] = best; s_i[wave] = bidx; }
    __syncthreads();

    if (wave == 0) {
        float v2 = s_v[lane];
        int   i2 = s_i[lane];
        #pragma unroll
        for (int off = 16; off > 0; off >>= 1) {
            const float ov = __shfl_xor(v2, off, 32);
            const int   oi = __shfl_xor(i2, off, 32);
            amax_combine(v2, i2, ov, oi);
        }
        if (lane == 0) {
            const int idx = i2;
            const int row = idx / LOC_W;
            const int col = idx - row * LOC_W;

            // bias_pos = [cr0*(H-1) - row, cr1*(W-1) - col]; h==w==192 makes
            // the reference's (w,h)-major flattening coincide with row*W+col.
            const float cr0 = center_rate[2 * b + 0];
            const float cr1 = center_rate[2 * b + 1];
            const float bias0 = cr0 * (float)(LOC_H - 1) - (float)row;
            const float bias1 = cr1 * (float)(LOC_W - 1) - (float)col;

            const float* lb = loc + (size_t)b * 2u * (size_t)hw;
            const float r0 = lb[idx];
            const float r1 = lb[hw + idx];

            const float d0 = fabsf(r0 - bias0);
            const float d1 = fabsf(r1 - bias1);
            const float l0 = (d0 < 1.0f) ? 0.5f * d0 * d0 : d0 - 0.5f;
            const float l1 = (d1 < 1.0f) ? 0.5f * d1 * d1 : d1 - 0.5f;
            per_sample[b] = l0 + l1;
        }
    }
}

// Fast path: hw == NITER*STEP exactly.  Fully unrolled uniform-trip scan:
// NITER clustered global_load_b128 per thread, and the scan-local index
// `code = k*4+j` is a compile-time constant, so each element costs a single
// v_cmp + two v_cndmask (inline-constant index operand).  Strict `>` keeps
// the first occurrence because the per-thread scan order is ascending-index.
template <int NITER>
__global__ __launch_bounds__(BLOCK_THREADS, 1)
void locloss_argmax_fixed(const float* __restrict__ cls,
                          const float* __restrict__ loc,
                          const float* __restrict__ center_rate,
                          float* __restrict__ per_sample,
                          int hw) {
    const int b   = blockIdx.x;
    const int tid = threadIdx.x;
    const float* base = cls + (size_t)b * (size_t)hw;

    float best = -INFINITY;
    int   code = 0;                    // k*4 + j (compile-time constants)

    #pragma unroll
    for (int k = 0; k < NITER; ++k) {
        const float4 v =
            *reinterpret_cast<const float4*>(base + (size_t)(k * BLOCK_THREADS + tid) * 4u);
        if (v.x > best) { best = v.x; code = k * 4 + 0; }
        if (v.y > best) { best = v.y; code = k * 4 + 1; }
        if (v.z > best) { best = v.z; code = k * 4 + 2; }
        if (v.w > best) { best = v.w; code = k * 4 + 3; }
    }

    const int bidx = (code >> 2) * STEP + tid * 4 + (code & 3);
    block_finish(best, bidx, b, hw, loc, center_rate, per_sample);
}

// Generic fallback: uniform scalar trip count + guarded scalar tail.
__global__ __launch_bounds__(BLOCK_THREADS, 1)
void locloss_argmax_generic(const float* __restrict__ cls,
                            const float* __restrict__ loc,
                            const float* __restrict__ center_rate,
                            float* __restrict__ per_sample,
                            int hw) {
    const int b   = blockIdx.x;
    const int tid = threadIdx.x;
    const float* base = cls + (size_t)b * (size_t)hw;

    float best = -INFINITY;
    int   bidx = 0x7FFFFFFF;

    const int niter = hw / STEP;       // uniform across the block
    for (int k = 0; k < niter; ++k) {
        const int e = (k * BLOCK_THREADS + tid) * 4;
        const float4 v = *reinterpret_cast<const float4*>(base + e);
        if (v.x > best) { best = v.x; bidx = e + 0; }
        if (v.y > best) { best = v.y; bidx = e + 1; }
        if (v.z > best) { best = v.z; bidx = e + 2; }
        if (v.w > best) { best = v.w; bidx = e + 3; }
    }
    // tail (indices all larger than the main part -> ascending order holds)
    for (int e = niter * STEP + tid; e < hw; e += BLOCK_THREADS) {
        const float v = base[e];
        if (v > best) { best = v; bidx = e; }
    }

    block_finish(best, bidx, b, hw, loc, center_rate, per_sample);
}

// Deterministic final reduction: one wave32, fixed strided accumulation order
// then a fixed xor tree.  No float atomics -> bit-identical across replays.
__global__ void locloss_reduce_kernel(const float* __restrict__ per_sample,
                                      float* __restrict__ out,
                                      int b) {
    const int lane = threadIdx.x;      // 32 threads
    float s = 0.0f;
    for (int i = lane; i < b; i += 32) s += per_sample[i];
    #pragma unroll
    for (int off = 16; off > 0; off >>= 1) s += __shfl_xor(s, off, 32);
    if (lane == 0) out[0] = s / (float)(2 * b);
}

extern "C" void kernel_launch(void* const* d_in, const int* in_sizes, int n_in,
                              void* d_out, int out_size, void* d_ws, size_t ws_size,
                              hipStream_t stream) {
    const float* cls = (const float*)d_in[0];  // (b, 1, h, w) f32
    const float* loc = (const float*)d_in[1];  // (b, 2, h, w) f32
    const float* cr  = (const float*)d_in[2];  // (b, 2)       f32

    const int b  = in_sizes[2] / 2;            // 256
    const int hw = in_sizes[0] / b;            // 36864 = 9 * STEP

    float* per_sample = (float*)d_ws;          // b floats of scratch

    if (hw == 9 * STEP) {
        locloss_argmax_fixed<9><<<b, BLOCK_THREADS, 0, stream>>>(
            cls, loc, cr, per_sample, hw);
    } else {
        locloss_argmax_generic<<<b, BLOCK_THREADS, 0, stream>>>(
            cls, loc, cr, per_sample, hw);
    }
    locloss_reduce_kernel<<<1, 32, 0, stream>>>(per_sample, (float*)d_out, b);
}